// SimpleGNN_76510547411600
// MI455X (gfx1250) — compile-verified
//
#include <hip/hip_runtime.h>
#include <hip/hip_bf16.h>

typedef __attribute__((ext_vector_type(2))) float v2f;
typedef __attribute__((ext_vector_type(8))) float v8f;

// no-return global atomic add: GLOBAL_ATOMIC_ADD_F32 (STOREcnt path, no flat aperture check)
__device__ __forceinline__ void gatomic_add4(float* o, float a, float b, float c, float d) {
    asm volatile("global_atomic_add_f32 %0, %1, off\n\t"
                 "global_atomic_add_f32 %0, %2, off offset:4\n\t"
                 "global_atomic_add_f32 %0, %3, off offset:8\n\t"
                 "global_atomic_add_f32 %0, %4, off offset:12"
                 :: "v"(o), "v"(a), "v"(b), "v"(c), "v"(d) : "memory");
}
__device__ __forceinline__ void gatomic_add1(float* o, float a) {
    asm volatile("global_atomic_add_f32 %0, %1, off" :: "v"(o), "v"(a) : "memory");
}

// ---------------- generic zero ----------------
__global__ void zero_f32(float* __restrict__ p, long long n4) {
    long long i = blockIdx.x * (long long)blockDim.x + threadIdx.x;
    long long stride = (long long)gridDim.x * blockDim.x;
    float4 z = {0.f, 0.f, 0.f, 0.f};
    for (; i < n4; i += stride) ((float4*)p)[i] = z;
}

// ---------------- degree + laplacian norm ----------------
__global__ void deg_kernel(const long long* __restrict__ row, float* __restrict__ deg, long long E) {
    long long e = blockIdx.x * (long long)blockDim.x + threadIdx.x;
    if (e < E) gatomic_add1(&deg[row[e]], 1.0f);
}

__global__ void norm_kernel(const long long* __restrict__ row, const long long* __restrict__ col,
                            const float* __restrict__ deg, float* __restrict__ norm, long long E) {
    long long e = blockIdx.x * (long long)blockDim.x + threadIdx.x;
    if (e < E) {
        float dr = deg[row[e]], dc = deg[col[e]];
        float ir = dr > 0.f ? rsqrtf(fmaxf(dr, 1e-12f)) : 0.f;
        float ic = dc > 0.f ? rsqrtf(fmaxf(dc, 1e-12f)) : 0.f;
        norm[e] = -ir * ic;
    }
}

// ---------------- sparse propagation: out[row] += norm * z[col] ----------------
// 16 threads per edge, float4 per thread -> coalesced 256B per edge row.
__global__ void prop_kernel(const float* __restrict__ z, const float* __restrict__ norm,
                            const long long* __restrict__ row, const long long* __restrict__ col,
                            float* __restrict__ out, long long E) {
    long long t = blockIdx.x * (long long)blockDim.x + threadIdx.x;
    long long nwork = E * 16;
    if (t >= nwork) return;
    long long e = t >> 4;
    int g = (int)(t & 15);
    if (g == 0) {   // stream-ahead prefetch of edge index/weight arrays (global_prefetch_b8)
        __builtin_prefetch(col + e + 4096, 0, 1);
        __builtin_prefetch(row + e + 4096, 0, 1);
        __builtin_prefetch(norm + e + 8192, 0, 1);
    }
    float w = norm[e];
    long long c = col[e], r = row[e];
    float4 zv = *(const float4*)(z + c * 64 + g * 4);
    float* o = out + r * 64 + g * 4;
    gatomic_add4(o, w * zv.x, w * zv.y, w * zv.z, w * zv.w);
}

// ---------------- combine Cheb weights: [W0-W2 ; W1 ; 2*W2] (192x64) ----------------
__global__ void build_wcomb(const float* __restrict__ W, float* __restrict__ wc) {
    int i = blockIdx.x * blockDim.x + threadIdx.x;
    if (i < 4096)           wc[i] = W[i] - W[2 * 4096 + i];
    else if (i < 2 * 4096)  wc[i] = W[i];
    else if (i < 3 * 4096)  wc[i] = 2.0f * W[i];
}

// ---------------- fused multi-source GEMM via V_WMMA_F32_16X16X4_F32 ----------------
// out[n,64] = sum_s S_s @ W[s*64 .. s*64+63, :]  + bias
// One wave computes a 16-row x 64-col tile (4x 16x16 WMMA accumulators).
// Weights staged in LDS K-paired: wlds[(k>>1)*128 + col*2 + (k&1)] so a B fragment
// is one contiguous ds_load_b64 into an even-aligned WMMA source pair.
__global__ __launch_bounds__(256) void gemm_multi(
    const float* __restrict__ S0, const float* __restrict__ S1, const float* __restrict__ S2,
    const float* __restrict__ W, const float* __restrict__ bias,
    float* __restrict__ out, int n, int nsrc)
{
    __shared__ float wlds[192 * 64];  // 48 KB
    int krows = nsrc * 64;
    for (int i = threadIdx.x; i < krows * 64; i += blockDim.x) {
        int k = i >> 6, c = i & 63;
        wlds[((k >> 1) * 64 + c) * 2 + (k & 1)] = W[i];
    }
    __syncthreads();

    int wave = threadIdx.x >> 5;
    int lane = threadIdx.x & 31;
    int tile = blockIdx.x * 8 + wave;
    int row0 = tile * 16;
    if (row0 >= n) return;                 // wave-uniform exit: EXEC stays all-ones

    int m  = lane & 15;                    // A: row within tile / B,C: column within tile
    int hi = lane >> 4;                    // half-wave selector
    int arow = row0 + m;
    if (arow >= n) arow = n - 1;           // clamp (keeps EXEC full)

    v8f acc0 = {}, acc1 = {}, acc2 = {}, acc3 = {};

    // One pass per source, pointer used directly (global_load, not flat).
    auto pass = [&](const float* __restrict__ S, int sbase) {
        const float* aptr = S + (size_t)arow * 64 + hi * 2;
#pragma unroll
        for (int k0 = 0; k0 < 64; k0 += 4) {
            // A 16x4 f32 frag: v0 = K=(0|2), v1 = K=(1|3) depending on half-wave
            float2 a = *(const float2*)(aptr + k0);
            v2f av = {a.x, a.y};
            int kk = sbase + k0 + hi * 2;                // even
            const float* bbase = wlds + kk * 64 + m * 2; // paired layout
#pragma unroll
            for (int t = 0; t < 4; ++t) {
                v2f bv = *(const v2f*)(bbase + t * 32);
                v8f* accp = (t == 0) ? &acc0 : (t == 1) ? &acc1 : (t == 2) ? &acc2 : &acc3;
                *accp = __builtin_amdgcn_wmma_f32_16x16x4_f32(
                    false, av, false, bv, (short)0, *accp, false, false);
            }
        }
    };
    pass(S0, 0);
    if (nsrc > 1) pass(S1, 64);
    if (nsrc > 2) pass(S2, 128);

    // C layout: VGPR v -> M = v + 8*hi, N = lane&15 (per 16x16 tile)
    v8f accs[4] = {acc0, acc1, acc2, acc3};
#pragma unroll
    for (int t = 0; t < 4; ++t) {
        int col = t * 16 + m;
        float bv = bias ? bias[col] : 0.f;
#pragma unroll
        for (int v = 0; v < 8; ++v) {
            int r = row0 + v + hi * 8;
            if (r < n) out[(size_t)r * 64 + col] = accs[t][v] + bv;
        }
    }
}

// ---------------- BatchNorm stats: per-column sum & sumsq ----------------
__global__ __launch_bounds__(256) void bn_stats(const float* __restrict__ x,
                                                float* __restrict__ stats, int n) {
    int c = threadIdx.x & 63;
    int grp = (blockIdx.x * blockDim.x + threadIdx.x) >> 6;
    int stride = (gridDim.x * blockDim.x) >> 6;
    float s = 0.f, s2 = 0.f;
    for (int r = grp; r < n; r += stride) {
        float v = x[(size_t)r * 64 + c];
        s += v; s2 += v * v;
    }
    __shared__ float ls[256], ls2[256];
    ls[threadIdx.x] = s; ls2[threadIdx.x] = s2;
    __syncthreads();
    if (threadIdx.x < 64) {
        s  = ls[threadIdx.x]  + ls[threadIdx.x + 64]  + ls[threadIdx.x + 128]  + ls[threadIdx.x + 192];
        s2 = ls2[threadIdx.x] + ls2[threadIdx.x + 64] + ls2[threadIdx.x + 128] + ls2[threadIdx.x + 192];
        gatomic_add1(&stats[c], s);
        gatomic_add1(&stats[64 + c], s2);
    }
}

__global__ void bn_coef(const float* __restrict__ stats, const float* __restrict__ g,
                        const float* __restrict__ be, float* __restrict__ coef, int n) {
    int c = threadIdx.x;
    if (c < 64) {
        float inv = 1.0f / (float)n;
        float mu  = stats[c] * inv;
        float var = fmaxf(stats[64 + c] * inv - mu * mu, 0.f);
        float a   = g[c] * rsqrtf(var + 1e-5f);
        coef[c]      = a;
        coef[64 + c] = be[c] - mu * a;
    }
}

// ---------------- fused BN affine + leakyReLU (+ optional shortcut add) ----------------
__global__ void bn_apply(const float* __restrict__ x, const float* __restrict__ coef,
                         const float* __restrict__ sc, float* __restrict__ out, long long n4) {
    long long i = blockIdx.x * (long long)blockDim.x + threadIdx.x;
    if (i >= n4) return;
    int cb = (int)((i * 4) & 63);
    float4 a = *(const float4*)(coef + cb);
    float4 b = *(const float4*)(coef + 64 + cb);
    float4 v = ((const float4*)x)[i];
    float4 y;
    y.x = fmaf(a.x, v.x, b.x); y.y = fmaf(a.y, v.y, b.y);
    y.z = fmaf(a.z, v.z, b.z); y.w = fmaf(a.w, v.w, b.w);
    y.x = y.x > 0.f ? y.x : 0.01f * y.x;
    y.y = y.y > 0.f ? y.y : 0.01f * y.y;
    y.z = y.z > 0.f ? y.z : 0.01f * y.z;
    y.w = y.w > 0.f ? y.w : 0.01f * y.w;
    if (sc) {
        float4 s = ((const float4*)sc)[i];
        y.x += s.x; y.y += s.y; y.z += s.z; y.w += s.w;
    }
    ((float4*)out)[i] = y;
}

// ---------------- segment-max pooling via ordered-uint atomicMax ----------------
__global__ void pool_max(const float* __restrict__ h, const long long* __restrict__ batch,
                         unsigned* __restrict__ pooled, long long total) {
    long long t = blockIdx.x * (long long)blockDim.x + threadIdx.x;
    if (t >= total) return;
    long long r = t >> 6;
    int c = (int)(t & 63);
    int g = (int)batch[r];
    unsigned bits = __float_as_uint(h[t]);
    unsigned key = (bits & 0x80000000u) ? ~bits : (bits | 0x80000000u);
    atomicMax(&pooled[g * 64 + c], key);
}

__global__ void final_lin(const unsigned* __restrict__ pooled, const float* __restrict__ w_lin,
                          const float* __restrict__ b_lin, float* __restrict__ out, int G) {
    int g = threadIdx.x;
    if (g < G) {
        float acc = b_lin[0];
        for (int c = 0; c < 64; ++c) {
            unsigned key = pooled[g * 64 + c];
            unsigned bits = (key & 0x80000000u) ? (key ^ 0x80000000u) : ~key;
            acc = fmaf(__uint_as_float(bits), w_lin[c], acc);
        }
        out[g] = acc;
    }
}

// ---------------- host orchestration ----------------
extern "C" void kernel_launch(void* const* d_in, const int* in_sizes, int n_in,
                              void* d_out, int out_size, void* d_ws, size_t ws_size,
                              hipStream_t stream) {
    const float*     x     = (const float*)d_in[0];
    const long long* ei    = (const long long*)d_in[1];
    const long long* batch = (const long long*)d_in[2];
    const float* w1 = (const float*)d_in[3],  *b1 = (const float*)d_in[4];
    const float* w2 = (const float*)d_in[5],  *b2 = (const float*)d_in[6];
    const float* w3 = (const float*)d_in[7],  *b3 = (const float*)d_in[8];
    const float* g1 = (const float*)d_in[9],  *be1 = (const float*)d_in[10];
    const float* g2 = (const float*)d_in[11], *be2 = (const float*)d_in[12];
    const float* g3 = (const float*)d_in[13], *be3 = (const float*)d_in[14];
    const float* w_sc  = (const float*)d_in[15], *b_sc  = (const float*)d_in[16];
    const float* w_lin = (const float*)d_in[17], *b_lin = (const float*)d_in[18];

    const long long N = in_sizes[0] / 64;
    const long long E = in_sizes[1] / 2;
    const int G = out_size;                // [G,1] output
    const long long N64 = N * 64;

    float* ws = (float*)d_ws;
    size_t off = 0;
    auto alloc = [&](size_t nf) { size_t p = off; off += (nf + 63) & ~(size_t)63; return p; };
    float* deg    = ws + alloc((size_t)N);
    float* norm   = ws + alloc((size_t)E);
    float* wcomb  = ws + alloc(192 * 64);
    float* stats  = ws + alloc(128);
    float* coef   = ws + alloc(128);
    float* bufA   = ws + alloc((size_t)N64);
    float* bufB   = ws + alloc((size_t)N64);
    float* bufC   = ws + alloc((size_t)N64);
    float* bufD   = ws + alloc((size_t)N64);
    unsigned* pooled = (unsigned*)(ws + alloc((size_t)G * 64));
    (void)ws_size; (void)n_in;

    const long long* erow = ei;
    const long long* ecol = ei + E;

    auto zero = [&](float* p, long long nf) {
        long long n4 = nf / 4;
        int blocks = (int)((n4 + 255) / 256);
        if (blocks > 16384) blocks = 16384;
        zero_f32<<<blocks, 256, 0, stream>>>(p, n4);
    };
    auto prop = [&](const float* z, float* dst) {
        zero(dst, N64);
        long long nwork = E * 16;
        prop_kernel<<<(int)((nwork + 255) / 256), 256, 0, stream>>>(z, norm, erow, ecol, dst, E);
    };
    auto gemm = [&](const float* S0, const float* S1, const float* S2,
                    const float* W, const float* bias, float* outp, int nsrc) {
        int tiles = (int)((N + 15) / 16);
        gemm_multi<<<(tiles + 7) / 8, 256, 0, stream>>>(S0, S1, S2, W, bias, outp, (int)N, nsrc);
    };
    auto bn = [&](const float* conv, const float* g, const float* be, const float* sc, float* outp) {
        zero(stats, 128);
        bn_stats<<<512, 256, 0, stream>>>(conv, stats, (int)N);
        bn_coef<<<1, 64, 0, stream>>>(stats, g, be, coef, (int)N);
        long long n4 = N64 / 4;
        bn_apply<<<(int)((n4 + 255) / 256), 256, 0, stream>>>(conv, coef, sc, outp, n4);
    };

    // ---- graph normalization ----
    zero(deg, N);
    deg_kernel<<<(int)((E + 255) / 256), 256, 0, stream>>>(erow, deg, E);
    norm_kernel<<<(int)((E + 255) / 256), 256, 0, stream>>>(erow, ecol, deg, norm, E);

    // ---- layer 1 (Tx0 = x) ----
    prop(x, bufA);                 // Tx1
    prop(bufA, bufB);              // P
    build_wcomb<<<48, 256, 0, stream>>>(w1, wcomb);
    gemm(x, bufA, bufB, wcomb, b1, bufC, 3);
    bn(bufC, g1, be1, nullptr, bufA);          // h1 -> A

    // ---- layer 2 (Tx0 = A) ----
    prop(bufA, bufB);
    prop(bufB, bufC);
    build_wcomb<<<48, 256, 0, stream>>>(w2, wcomb);
    gemm(bufA, bufB, bufC, wcomb, b2, bufD, 3);
    bn(bufD, g2, be2, nullptr, bufB);          // h2 -> B

    // ---- layer 3 (Tx0 = B) ----
    prop(bufB, bufC);
    prop(bufC, bufD);
    build_wcomb<<<48, 256, 0, stream>>>(w3, wcomb);
    gemm(bufB, bufC, bufD, wcomb, b3, bufA, 3);          // conv3 -> A
    gemm(x, nullptr, nullptr, w_sc, b_sc, bufC, 1);      // shortcut -> C
    bn(bufA, g3, be3, bufC, bufD);                       // h3 = lrelu(bn(conv3)) + sc -> D

    // ---- pooling + final linear ----
    zero((float*)pooled, (long long)G * 64);
    pool_max<<<(int)((N64 + 255) / 256), 256, 0, stream>>>(bufD, batch, pooled, N64);
    final_lin<<<1, ((G + 63) / 64) * 64, 0, stream>>>(pooled, w_lin, b_lin, (float*)d_out, G);
}